// MultitaskReadout_2542620639496
// MI455X (gfx1250) — compile-verified
//
#include <hip/hip_runtime.h>

typedef __attribute__((ext_vector_type(2))) float v2f;
typedef __attribute__((ext_vector_type(8))) float v8f;

#define D_LAT      1024
#define N_OUT      34
#define N_PAD      48             // real packed W rows (3 N-tiles of 16)
#define W_ROWS     64             // staged rows (48 real + 16 pad -> uniform async counts)
#define KC         32             // K-chunk per stage
#define NCHUNK     (D_LAT / KC)   // 32
#define STRIDE     36             // floats: 144B rows -> 16B aligned, bank-conflict-free
#define TOK_PER_WG 128
#define NT_TOTAL   (16 * 4096)

#define XBUF_FLOATS (TOK_PER_WG * STRIDE)   // 4608
#define WBUF_FLOATS (W_ROWS * STRIDE)       // 2304

// ---------------------------------------------------------------------------
// Prep: pack W0/W1/W2 -> Wcat[64][1024] (rows 34..63 zero), b* -> bcat[48]
// ---------------------------------------------------------------------------
__global__ void pack_weights_kernel(const float* __restrict__ W0, const float* __restrict__ b0,
                                    const float* __restrict__ W1, const float* __restrict__ b1,
                                    const float* __restrict__ W2, const float* __restrict__ b2,
                                    float* __restrict__ Wcat, float* __restrict__ bcat) {
    int i = blockIdx.x * 256 + threadIdx.x;
    if (i < W_ROWS * D_LAT) {
        int row = i >> 10;
        int k   = i & (D_LAT - 1);
        float v = 0.0f;
        if (row < 2)       v = W0[row * D_LAT + k];
        else if (row < 4)  v = W1[(row - 2) * D_LAT + k];
        else if (row < 34) v = W2[(row - 4) * D_LAT + k];
        Wcat[i] = v;
    }
    if (i < N_PAD) {
        float v = 0.0f;
        if (i < 2)       v = b0[i];
        else if (i < 4)  v = b1[i - 2];
        else if (i < 34) v = b2[i - 4];
        bcat[i] = v;
    }
}

// ---------------------------------------------------------------------------
// Async global -> LDS b128 (gfx1250 ASYNCcnt path, bypasses VGPRs)
// ---------------------------------------------------------------------------
__device__ __forceinline__ void async_ld_b128(unsigned lds_byte_off, const float* gaddr) {
    asm volatile("global_load_async_to_lds_b128 %0, %1, off"
                 :: "v"(lds_byte_off), "v"(gaddr)
                 : "memory");
}

// ---------------------------------------------------------------------------
// Main: 8 waves/block, 16 tokens/wave, N=48 via 3 f32-WMMA tiles,
//       double-buffered async-DMA staging.
// ---------------------------------------------------------------------------
__global__ __launch_bounds__(256) void multitask_readout_kernel(
        const float* __restrict__ x, const int* __restrict__ didx,
        const float* __restrict__ Wcat, const float* __restrict__ bcat,
        float* __restrict__ out) {
    __shared__ float sX[2 * XBUF_FLOATS];   // 36,864 B
    __shared__ float sW[2 * WBUF_FLOATS];   // 18,432 B

    const int tid   = threadIdx.x;
    const int lane  = tid & 31;
    const int wave  = tid >> 5;
    const int wgTok = blockIdx.x * TOK_PER_WG;

    const int lrow    = lane & 15;          // M (A) / N (B) within tile
    const int khalf   = (lane >> 4) * 2;    // lanes 0-15: K+0..1, lanes 16-31: K+2..3
    const int waveTok = wave * 16;

    // LDS byte offsets of the buffers (generic LDS addr: low 32 bits = LDS offset)
    const unsigned sxOff = (unsigned)(unsigned long long)(const void*)sX;
    const unsigned swOff = (unsigned)(unsigned long long)(const void*)sW;

    // Stage one K-chunk into LDS buffer `buf`: exactly 6 async b128 per thread.
    auto stage = [&](int c, int buf) {
        const int kbase = c * KC;
        // ---- X: 128 rows x 32 floats = 1024 b128 slots, 4 per thread ----
        {
            const int r   = tid >> 1;           // 0..127
            const int sub = tid & 1;            // low/high 64B of the row chunk
            const float* g = x + (size_t)(wgTok + r) * D_LAT + kbase + sub * 16;
            const unsigned l = sxOff + (unsigned)(buf * XBUF_FLOATS + r * STRIDE + sub * 16) * 4u;
#pragma unroll
            for (int p = 0; p < 4; ++p)
                async_ld_b128(l + p * 16u, g + p * 4);
        }
        // ---- W: 64 rows x 32 floats = 512 b128 slots, 2 per thread ----
        {
#pragma unroll
            for (int q = 0; q < 2; ++q) {
                const int s    = tid * 2 + q;
                const int row  = s >> 3;        // 0..63
                const int slot = s & 7;
                const float* g = Wcat + (size_t)row * D_LAT + kbase + slot * 4;
                const unsigned l = swOff + (unsigned)(buf * WBUF_FLOATS + row * STRIDE + slot * 4) * 4u;
                async_ld_b128(l, g);
            }
        }
    };

    v8f acc0 = {}, acc1 = {}, acc2 = {};

    stage(0, 0);

#pragma unroll 1
    for (int c = 0; c < NCHUNK; ++c) {
        if (c + 1 < NCHUNK) {
            stage(c + 1, (c + 1) & 1);                      // DMA next chunk (other buffer)
            asm volatile("s_wait_asynccnt 0x6" ::: "memory"); // stage c's 6 ops are done
        } else {
            asm volatile("s_wait_asynccnt 0x0" ::: "memory");
        }
        __syncthreads();   // all threads' DMA for chunk c visible

        const int buf = c & 1;
        const float* aB  = sX + buf * XBUF_FLOATS + (waveTok + lrow) * STRIDE + khalf;
        const float* w0B = sW + buf * WBUF_FLOATS + (0 * 16 + lrow) * STRIDE + khalf;
        const float* w1B = sW + buf * WBUF_FLOATS + (1 * 16 + lrow) * STRIDE + khalf;
        const float* w2B = sW + buf * WBUF_FLOATS + (2 * 16 + lrow) * STRIDE + khalf;

#pragma unroll
        for (int kk = 0; kk < KC; kk += 4) {
            v2f a  = *(const v2f*)(aB  + kk);
            v2f w0 = *(const v2f*)(w0B + kk);
            v2f w1 = *(const v2f*)(w1B + kk);
            v2f w2 = *(const v2f*)(w2B + kk);
            acc0 = __builtin_amdgcn_wmma_f32_16x16x4_f32(false, a, false, w0, (short)0, acc0, false, false);
            acc1 = __builtin_amdgcn_wmma_f32_16x16x4_f32(false, a, false, w1, (short)0, acc1, false, false);
            acc2 = __builtin_amdgcn_wmma_f32_16x16x4_f32(false, a, false, w2, (short)0, acc2, false, false);
        }
        __syncthreads();   // everyone done reading buf before it is re-DMA'd
    }

    // ---- epilogue: bias + decoder mask + scattered b32 stores ----
    // C layout: VGPR j, lanes 0-15 -> M=j ; lanes 16-31 -> M=j+8 ; N = lane&15
    const int tokBase = wgTok + waveTok + (lane >> 4) * 8;
    int di[8];
#pragma unroll
    for (int j = 0; j < 8; ++j) di[j] = didx[tokBase + j];

#pragma unroll
    for (int nt = 0; nt < 3; ++nt) {
        const int col = nt * 16 + lrow;
        if (col < N_OUT) {
            const float bias = bcat[col];
            const int dec = (col < 2) ? 0 : (col < 4) ? 1 : 2;
            v8f acc = (nt == 0) ? acc0 : (nt == 1) ? acc1 : acc2;
#pragma unroll
            for (int j = 0; j < 8; ++j) {
                const float v = (di[j] == dec) ? (acc[j] + bias) : 0.0f;
                out[(size_t)(tokBase + j) * N_OUT + col] = v;
            }
        }
    }
}

// ---------------------------------------------------------------------------
extern "C" void kernel_launch(void* const* d_in, const int* in_sizes, int n_in,
                              void* d_out, int out_size, void* d_ws, size_t ws_size,
                              hipStream_t stream) {
    const float* x    = (const float*)d_in[0];
    const int*   didx = (const int*)d_in[1];
    const float* W0   = (const float*)d_in[2];
    const float* b0   = (const float*)d_in[3];
    const float* W1   = (const float*)d_in[4];
    const float* b1   = (const float*)d_in[5];
    const float* W2   = (const float*)d_in[6];
    const float* b2   = (const float*)d_in[7];
    float* out = (float*)d_out;

    float* Wcat = (float*)d_ws;                 // 64*1024 floats (rows 34..63 zero)
    float* bcat = Wcat + W_ROWS * D_LAT;        // 48 floats

    pack_weights_kernel<<<(W_ROWS * D_LAT + 255) / 256, 256, 0, stream>>>(
        W0, b0, W1, b1, W2, b2, Wcat, bcat);

    multitask_readout_kernel<<<NT_TOTAL / TOK_PER_WG, 256, 0, stream>>>(
        x, didx, Wcat, bcat, out);
}